// ContrastiveLoss_23364622090316
// MI455X (gfx1250) — compile-verified
//
#include <hip/hip_runtime.h>
#include <math.h>

// ---- problem dims (fixed by the reference harness) ----
#define B_IMG 128
#define R_REG 36
#define C_CAP 128
#define W_WORD 32
#define D_FEAT 256
#define NPAD 48   // regions padded to 3 WMMA n-tiles

typedef __attribute__((ext_vector_type(2))) float v2f;
typedef __attribute__((ext_vector_type(8))) float v8f;

static __device__ __forceinline__ v8f wmma_f32(v2f a, v2f b, v8f c) {
  // V_WMMA_F32_16X16X4_F32 : D = A(16x4 f32) * B(4x16 f32) + C(16x16 f32)
  return __builtin_amdgcn_wmma_f32_16x16x4_f32(false, a, false, b, (short)0, c,
                                               false, false);
}

static __device__ __forceinline__ float wave_max32(float v) {
#pragma unroll
  for (int off = 16; off > 0; off >>= 1) v = fmaxf(v, __shfl_xor(v, off, 32));
  return v;
}
static __device__ __forceinline__ float wave_sum32(float v) {
#pragma unroll
  for (int off = 16; off > 0; off >>= 1) v += __shfl_xor(v, off, 32);
  return v;
}

// ---------------------------------------------------------------------------
// Prep 1: per-region image L2 norms  (B*R rows of D floats)
// ---------------------------------------------------------------------------
__global__ __launch_bounds__(256) void imnorm_kernel(
    const float* __restrict__ im, float* __restrict__ nrm) {
  const int row = blockIdx.x * 8 + (threadIdx.x >> 5);
  const int lane = threadIdx.x & 31;
  if (row >= B_IMG * R_REG) return;
  float ss = 0.f;
  for (int d = lane; d < D_FEAT; d += 32) {
    float v = im[(size_t)row * D_FEAT + d];
    ss += v * v;
  }
  ss = wave_sum32(ss);
  if (lane == 0) nrm[row] = sqrtf(ss);
}

// ---------------------------------------------------------------------------
// Prep 2: per-caption word Gram matrices G_c = S_c * S_c^T (32x32), via WMMA.
// One wave per caption. B-fragment of S^T is lane-identical to the A-fragment.
// ---------------------------------------------------------------------------
__global__ __launch_bounds__(32) void gram_kernel(
    const float* __restrict__ s, float* __restrict__ G) {
  const int c = blockIdx.x;
  const int lane = threadIdx.x;
  const int hi = lane >> 4;      // 0: K = kb+0..1, 1: K = kb+2..3
  const int lr = lane & 15;      // row/col within tile
  const float* Sc = s + (size_t)c * W_WORD * D_FEAT;

  const v8f zero8 = {0.f, 0.f, 0.f, 0.f, 0.f, 0.f, 0.f, 0.f};
  v8f acc[2][2];
#pragma unroll
  for (int mt = 0; mt < 2; ++mt)
#pragma unroll
    for (int nt = 0; nt < 2; ++nt) acc[mt][nt] = zero8;

#pragma unroll 2
  for (int kb = 0; kb < D_FEAT; kb += 4) {
    const int ko = kb + hi * 2;
    v2f a0 = *(const v2f*)(Sc + (size_t)(lr)*D_FEAT + ko);
    v2f a1 = *(const v2f*)(Sc + (size_t)(16 + lr) * D_FEAT + ko);
    acc[0][0] = wmma_f32(a0, a0, acc[0][0]);
    acc[0][1] = wmma_f32(a0, a1, acc[0][1]);
    acc[1][0] = wmma_f32(a1, a0, acc[1][0]);
    acc[1][1] = wmma_f32(a1, a1, acc[1][1]);
  }

  float* Gc = G + (size_t)c * W_WORD * W_WORD;
#pragma unroll
  for (int mt = 0; mt < 2; ++mt)
#pragma unroll
    for (int nt = 0; nt < 2; ++nt)
#pragma unroll
      for (int i = 0; i < 8; ++i)
        Gc[(mt * 16 + i + hi * 8) * W_WORD + nt * 16 + lr] = acc[mt][nt][i];
}

// ---------------------------------------------------------------------------
// Main fused kernel: one wave per (image b, caption c) pair.
//  - 32x36 raw dot tile via v_wmma_f32_16x16x4_f32 (K=256)
//  - leaky_relu + l2norm over regions, masked softmax over words
//  - num = 0.7 * sum_w A*dot_raw ; ||wC|| = 0.7*sqrt(a^T G a)
//  - LogSumExp over regions -> scores[b][c]
// ---------------------------------------------------------------------------
__global__ __launch_bounds__(32) void xattn_kernel(
    const float* __restrict__ im, const float* __restrict__ s,
    const int* __restrict__ s_l, const float* __restrict__ imnorm,
    const float* __restrict__ G, float* __restrict__ scores) {
  __shared__ float Traw[W_WORD][NPAD + 1];  // raw dots (stride 49: no bank conflicts)
  __shared__ float Awt[W_WORD][NPAD + 1];   // normalized attn, then softmax weights
  __shared__ float Gs[W_WORD][W_WORD];      // caption Gram matrix

  const int b = blockIdx.x;
  const int c = blockIdx.y;
  const int lane = threadIdx.x;
  const int hi = lane >> 4;
  const int lr = lane & 15;

  for (int i = lane; i < W_WORD * W_WORD; i += 32)
    Gs[i >> 5][i & 31] = G[(size_t)c * W_WORD * W_WORD + i];

  const float* Sc = s + (size_t)c * W_WORD * D_FEAT;
  const float* Ib = im + (size_t)b * R_REG * D_FEAT;

  const v8f zero8 = {0.f, 0.f, 0.f, 0.f, 0.f, 0.f, 0.f, 0.f};
  const v2f zero2 = {0.f, 0.f};
  v8f acc[2][3];
#pragma unroll
  for (int mt = 0; mt < 2; ++mt)
#pragma unroll
    for (int nt = 0; nt < 3; ++nt) acc[mt][nt] = zero8;

  const int r2 = 32 + lr;  // third n-tile region index (valid only lr<4)
#pragma unroll 2
  for (int kb = 0; kb < D_FEAT; kb += 4) {
    const int ko = kb + hi * 2;
    v2f a0 = *(const v2f*)(Sc + (size_t)(lr)*D_FEAT + ko);
    v2f a1 = *(const v2f*)(Sc + (size_t)(16 + lr) * D_FEAT + ko);
    v2f b0 = *(const v2f*)(Ib + (size_t)(lr)*D_FEAT + ko);
    v2f b1 = *(const v2f*)(Ib + (size_t)(16 + lr) * D_FEAT + ko);
    v2f b2 = zero2;
    if (r2 < R_REG) b2 = *(const v2f*)(Ib + (size_t)r2 * D_FEAT + ko);
    acc[0][0] = wmma_f32(a0, b0, acc[0][0]);
    acc[0][1] = wmma_f32(a0, b1, acc[0][1]);
    acc[0][2] = wmma_f32(a0, b2, acc[0][2]);
    acc[1][0] = wmma_f32(a1, b0, acc[1][0]);
    acc[1][1] = wmma_f32(a1, b1, acc[1][1]);
    acc[1][2] = wmma_f32(a1, b2, acc[1][2]);
  }

  // Spill C/D tiles: VGPR i holds M=i (lanes 0-15) and M=i+8 (lanes 16-31)
#pragma unroll
  for (int mt = 0; mt < 2; ++mt)
#pragma unroll
    for (int nt = 0; nt < 3; ++nt)
#pragma unroll
      for (int i = 0; i < 8; ++i)
        Traw[mt * 16 + i + hi * 8][nt * 16 + lr] = acc[mt][nt][i];
  __syncthreads();

  // Step 1: leaky_relu(0.1) then l2norm over regions (per word = lane)
  {
    const int w = lane;
    float ss = 0.f;
    for (int r = 0; r < R_REG; ++r) {
      float t = Traw[w][r];
      float lv = (t > 0.f) ? t : 0.1f * t;
      Awt[w][r] = lv;
      ss += lv * lv;
    }
    float inv = 1.f / (sqrtf(ss) + 1e-8f);
    for (int r = 0; r < R_REG; ++r) Awt[w][r] *= inv;
  }
  __syncthreads();

  const int len = s_l[c];

  // Step 2: masked softmax over words, per region column (in place in Awt)
  for (int r = lane; r < R_REG; r += 32) {
    float mx = -1e30f;
    for (int w = 0; w < W_WORD; ++w) {
      float lg = (w < len) ? 9.0f * Awt[w][r] : -1e9f;
      mx = fmaxf(mx, lg);
    }
    float ssum = 0.f;
    for (int w = 0; w < W_WORD; ++w)
      if (w < len) ssum += expf(9.0f * Awt[w][r] - mx);
    float invs = 1.f / ssum;
    for (int w = 0; w < W_WORD; ++w) {
      float e = (w < len) ? expf(9.0f * Awt[w][r] - mx) * invs : 0.f;
      Awt[w][r] = e;  // each entry read before overwritten
    }
  }
  __syncthreads();

  // Steps 3-5: cosine sim via Gram identity, then LSE over regions
  float lval[2] = {-1e30f, -1e30f};
#pragma unroll
  for (int p = 0; p < 2; ++p) {
    const int r = lane + 32 * p;
    if (r < R_REG) {
      float num = 0.f;
      for (int w = 0; w < W_WORD; ++w) num += Awt[w][r] * Traw[w][r];
      num *= 0.7f;  // WEIGHT_FACTOR
      float quad = 0.f;
      for (int w = 0; w < W_WORD; ++w) {
        float mw = 0.f;
        for (int w2 = 0; w2 < W_WORD; ++w2) mw += Gs[w][w2] * Awt[w2][r];
        quad += Awt[w][r] * mw;
      }
      quad = fmaxf(quad, 0.f);
      float wcn = 0.7f * sqrtf(quad);               // ||wC||
      float den = fmaxf(imnorm[b * R_REG + r] * wcn, 1e-8f);
      lval[p] = 6.0f * (num / den);                 // LAMBDA_LSE * row_sim
    }
  }
  float m = wave_max32(fmaxf(lval[0], lval[1]));
  float es = expf(lval[0] - m) + ((lval[1] > -1e29f) ? expf(lval[1] - m) : 0.f);
  es = wave_sum32(es);
  if (lane == 0) scores[b * C_CAP + c] = (logf(es) + m) / 6.0f;
}

// ---------------------------------------------------------------------------
// Hinge loss with hardest negatives over the 128x128 score matrix.
// ---------------------------------------------------------------------------
__global__ __launch_bounds__(128) void loss_kernel(const float* __restrict__ S,
                                                   float* __restrict__ out) {
  __shared__ float red[B_IMG];
  const int i = threadIdx.x;
  const float diag = S[i * C_CAP + i];
  float rowmax = 0.f, colmax = 0.f;
  for (int j = 0; j < C_CAP; ++j) {
    if (j == i) continue;
    float cs = 0.2f + S[i * C_CAP + j] - diag;   // cost_s[i][j], diag_i
    rowmax = fmaxf(rowmax, fmaxf(cs, 0.f));
    float ci = 0.2f + S[j * C_CAP + i] - diag;   // cost_im[j][i], diag of col i
    colmax = fmaxf(colmax, fmaxf(ci, 0.f));
  }
  red[i] = rowmax + colmax;
  __syncthreads();
  for (int off = 64; off > 0; off >>= 1) {
    if (i < off) red[i] += red[i + off];
    __syncthreads();
  }
  if (i == 0) out[0] = red[0];
}

// ---------------------------------------------------------------------------
extern "C" void kernel_launch(void* const* d_in, const int* in_sizes, int n_in,
                              void* d_out, int out_size, void* d_ws,
                              size_t ws_size, hipStream_t stream) {
  (void)in_sizes; (void)n_in; (void)out_size; (void)ws_size;
  const float* im = (const float*)d_in[0];   // (128, 36, 256) f32
  const float* s  = (const float*)d_in[1];   // (128, 32, 256) f32
  const int* s_l  = (const int*)d_in[2];     // (128,) i32
  float* out = (float*)d_out;                // scalar f32 loss

  float* scores = (float*)d_ws;                       // 128*128
  float* imnorm = scores + B_IMG * C_CAP;             // 128*36
  float* G      = imnorm + B_IMG * R_REG;             // 128*32*32

  imnorm_kernel<<<(B_IMG * R_REG + 7) / 8, 256, 0, stream>>>(im, imnorm);
  gram_kernel<<<C_CAP, 32, 0, stream>>>(s, G);
  dim3 grid(B_IMG, C_CAP);
  xattn_kernel<<<grid, 32, 0, stream>>>(im, s, s_l, imnorm, G, scores);
  loss_kernel<<<1, 128, 0, stream>>>(scores, out);
}